// NeuralFeatureGrid_19859928777079
// MI455X (gfx1250) — compile-verified
//
#include <hip/hip_runtime.h>
#include <hip/hip_bf16.h>
#include <math.h>

// ---------------------------------------------------------------------------
// Problem constants (match reference)
// ---------------------------------------------------------------------------
#define DIN    512          // decoder feature dim (LayerNorm + GEMM K)
#define NOUT   4096         // UF*UF*Z*FD
#define MROWS  2048         // B * DG*DG
#define NRAYS  16384        // B * R
#define NSAMP  128          // samples per ray
#define PRED_OFF    0
#define WEIGHT_OFF  16384                   // B*R
#define NFG_OFF     (16384 + 2097152)       // + B*R*S
#define NFG_ELEMS   8388608                 // 2*16*16*128*128

typedef __attribute__((ext_vector_type(2))) float v2f;
typedef __attribute__((ext_vector_type(8))) float v8f;

// ---------------------------------------------------------------------------
// CDNA5 async global->LDS copy (16B per lane), tracked by ASYNCcnt.
// lds_addr: LDS byte offset (low 32 bits of a generic pointer to __shared__).
// ---------------------------------------------------------------------------
__device__ __forceinline__ void async_copy_b128(unsigned lds_addr,
                                                const float* gptr) {
  asm volatile("global_load_async_to_lds_b128 %0, %1, off"
               :: "v"(lds_addr), "v"(gptr)
               : "memory");
}
__device__ __forceinline__ void wait_async0() {
  asm volatile("s_wait_asynccnt 0x0" ::: "memory");
}

// ---------------------------------------------------------------------------
// Kernel 1: per-row LayerNorm of decoder_output -> A_t (K-major, 512 x 2048)
// One wave32 per row; 8 waves per 256-thread block. K-major output makes the
// GEMM's A-tile an identity (async-copyable) 16B-chunked copy.
// ---------------------------------------------------------------------------
__global__ __launch_bounds__(256) void ln_kernel(const float* __restrict__ x,
                                                 const float* __restrict__ gam,
                                                 const float* __restrict__ bet,
                                                 float* __restrict__ out_t) {
  const int row  = blockIdx.x * 8 + (threadIdx.x >> 5);
  const int lane = threadIdx.x & 31;
  const float* xr = x + (size_t)row * DIN;
  float4 v[4];
  float s = 0.f, ss = 0.f;
#pragma unroll
  for (int i = 0; i < 4; ++i) {
    v[i] = *(const float4*)(xr + i * 128 + lane * 4);
    s  += v[i].x + v[i].y + v[i].z + v[i].w;
    ss += v[i].x * v[i].x + v[i].y * v[i].y + v[i].z * v[i].z + v[i].w * v[i].w;
  }
#pragma unroll
  for (int off = 16; off >= 1; off >>= 1) {
    s  += __shfl_xor(s, off, 32);
    ss += __shfl_xor(ss, off, 32);
  }
  const float mu   = s * (1.0f / DIN);
  const float var  = ss * (1.0f / DIN) - mu * mu;   // biased, matches reference
  const float rstd = rsqrtf(var + 1e-5f);
#pragma unroll
  for (int i = 0; i < 4; ++i) {
    const int k = i * 128 + lane * 4;
    const float* vv = (const float*)&v[i];
#pragma unroll
    for (int j = 0; j < 4; ++j)
      out_t[(size_t)(k + j) * MROWS + row] =
          (vv[j] - mu) * rstd * gam[k + j] + bet[k + j];
  }
}

// ---------------------------------------------------------------------------
// Kernel 2: WMMA fp32 GEMM  A_t^T(2048x512) x W1(512x4096) + b1, fused
// permute into nfg (B,F,Z,H,W) and gather-friendly nfg2 (B,Z,H,W,F).
// Block = 256 threads (8 waves) -> 32x64 output block; each wave owns a
// 16x16 tile via V_WMMA_F32_16X16X4_F32 (K step 4). Tiles are streamed into
// double-buffered LDS with GLOBAL_LOAD_ASYNC_TO_LDS_B128 so the copy of
// chunk k+1 overlaps the WMMAs of chunk k.
// ---------------------------------------------------------------------------
__global__ __launch_bounds__(256) void gemm_nfg_kernel(const float* __restrict__ At_g,
                                                       const float* __restrict__ Wm,
                                                       const float* __restrict__ bias,
                                                       float* __restrict__ nfg,
                                                       float* __restrict__ nfg2,
                                                       int use_nfg2) {
  __shared__ float At[2][64][32];   // K-major A tile, 2 x 8KB
  __shared__ float Bs[2][64][64];   // K x N B tile,   2 x 16KB

  const int t    = threadIdx.x;
  const int wave = t >> 5;
  const int lane = t & 31;
  const int half = lane >> 4;    // 0: lanes 0-15, 1: lanes 16-31
  const int mrow = lane & 15;
  const int wm   = wave >> 2;    // 0..1  (row tile within block)
  const int wn   = wave & 3;     // 0..3  (col tile within block)
  const int m0   = blockIdx.y * 32;
  const int n0   = blockIdx.x * 64;

  const unsigned baseA = (unsigned)(size_t)&At[0][0][0];
  const unsigned baseB = (unsigned)(size_t)&Bs[0][0][0];

  // Precompute per-thread chunk coordinates for the cooperative copies.
  // A tile: 64(K) x 32(M) floats = 512 b128 chunks, 2 per thread.
  const int ar0 = t >> 3,          ac0 = (t & 7) << 2;
  const int ar1 = (t + 256) >> 3,  ac1 = ((t + 256) & 7) << 2;
  // B tile: 64(K) x 64(N) floats = 1024 b128 chunks, 4 per thread.

  auto issue_tiles = [&](int buf, int k0) {
    const unsigned la = baseA + (unsigned)buf * 64 * 32 * 4;
    const unsigned lb = baseB + (unsigned)buf * 64 * 64 * 4;
    async_copy_b128(la + (unsigned)(ar0 * 32 + ac0) * 4,
                    At_g + (size_t)(k0 + ar0) * MROWS + m0 + ac0);
    async_copy_b128(la + (unsigned)(ar1 * 32 + ac1) * 4,
                    At_g + (size_t)(k0 + ar1) * MROWS + m0 + ac1);
#pragma unroll
    for (int i = 0; i < 4; ++i) {
      const int idx = t + i * 256;
      const int r = idx >> 4;
      const int c = (idx & 15) << 2;
      async_copy_b128(lb + (unsigned)(r * 64 + c) * 4,
                      Wm + (size_t)(k0 + r) * NOUT + n0 + c);
    }
  };

  // Prologue: fill buffer 0 with the first K chunk.
  issue_tiles(0, 0);
  wait_async0();
  __syncthreads();

  v8f acc = {};
  for (int k0 = 0; k0 < DIN; k0 += 64) {
    const int buf = (k0 >> 6) & 1;
    if (k0 + 64 < DIN)
      issue_tiles(buf ^ 1, k0 + 64);   // prefetch next chunk (overlaps WMMA)

#pragma unroll
    for (int kk = 0; kk < 64; kk += 4) {
      const int kb = kk + half * 2;    // half-wave owns K pair (ISA 7.12.2)
      v2f a, b;
      a[0] = At[buf][kb][wm * 16 + mrow];
      a[1] = At[buf][kb + 1][wm * 16 + mrow];
      b[0] = Bs[buf][kb][wn * 16 + mrow];
      b[1] = Bs[buf][kb + 1][wn * 16 + mrow];
      // 8 args: (neg_a, A, neg_b, B, c_mod, C, reuse_a, reuse_b)
      acc = __builtin_amdgcn_wmma_f32_16x16x4_f32(false, a, false, b,
                                                  (short)0, acc, false, false);
    }
    wait_async0();      // next chunk's async copies have landed in LDS
    __syncthreads();    // everyone done reading `buf`, copies visible
  }

  // Epilogue: bias + scatter into nfg (B,F,Z,H,W) and nfg2 (B,Z,H,W,F).
  // D layout: lane holds column N = lane%16; VGPR j holds M = j + 8*(lane/16).
  const int nloc = wn * 16 + mrow;
  const int col  = n0 + nloc;
  const int f  = col & 15;
  const int z  = (col >> 4) & 15;
  const int ux = (col >> 8) & 3;
  const int uy = (col >> 10) & 3;
  const float bv = bias[col];
#pragma unroll
  for (int j = 0; j < 8; ++j) {
    const int mloc = wm * 16 + half * 8 + j;
    const int row  = m0 + mloc;
    const int bb = row >> 10;
    const int g  = row & 1023;
    const int gy = g >> 5;
    const int gx = g & 31;
    const int h = gy * 4 + uy;
    const int w = gx * 4 + ux;
    const float val = acc[j] + bv;
    nfg[((((size_t)bb * 16 + f) * 16 + z) * 128 + h) * 128 + w] = val;
    if (use_nfg2)
      nfg2[((((size_t)bb * 16 + z) * 128 + h) * 128 + w) * 16 + f] = val;
  }
}

// ---------------------------------------------------------------------------
// Kernel 3: one 128-thread block per ray. Each thread = one depth sample:
// trilinear gather (16 feats) -> MLP 16->32->1 -> sigmoid -> alpha,
// LDS product-scan for transmittance, weights out, depth reduction.
// ---------------------------------------------------------------------------
__global__ __launch_bounds__(128) void ray_kernel(const float* __restrict__ ro,
                                                  const float* __restrict__ rd,
                                                  const float* __restrict__ W2,
                                                  const float* __restrict__ b2,
                                                  const float* __restrict__ W3,
                                                  const float* __restrict__ b3,
                                                  const float* __restrict__ nfg,
                                                  const float* __restrict__ nfg2,
                                                  int use_nfg2,
                                                  float* __restrict__ pred,
                                                  float* __restrict__ wout) {
  __shared__ float sW2[16 * 32];
  __shared__ float sb2[32];
  __shared__ float sW3[32];
  __shared__ float sb3;
  __shared__ float scan[NSAMP];
  __shared__ float red[NSAMP];

  const int t = threadIdx.x;
  for (int i = t; i < 512; i += 128) sW2[i] = W2[i];
  if (t < 32) { sb2[t] = b2[t]; sW3[t] = W3[t]; }
  if (t == 0) sb3 = b3[0];

  const int ray = blockIdx.x;
  const int bb  = ray >> 13;          // batch
  const float ox = ro[(size_t)ray * 3 + 0];
  const float oy = ro[(size_t)ray * 3 + 1];
  const float oz = ro[(size_t)ray * 3 + 2];
  const float dx = rd[(size_t)ray * 3 + 0];
  const float dy = rd[(size_t)ray * 3 + 1];
  const float dz = rd[(size_t)ray * 3 + 2];

  const float depth = 1.0f + 49.0f * ((float)t * (1.0f / 127.0f));
  const float px = ox + dx * depth;
  const float py = oy + dy * depth;
  const float pz = oz + dz * depth;
  const float xn = (px + 51.2f) * (2.0f / 102.4f) - 1.0f;
  const float yn = (py + 51.2f) * (2.0f / 102.4f) - 1.0f;
  const float zn = (pz + 5.0f)  * (2.0f / 8.0f)   - 1.0f;
  // grid last dim (x,y,z) -> (W,H,D); grid stacked [y,x,z]
  const float fw = (yn + 1.0f) * 0.5f * 127.0f;
  const float fh = (xn + 1.0f) * 0.5f * 127.0f;
  const float fd = (zn + 1.0f) * 0.5f * 15.0f;
  const float w0f = floorf(fw), h0f = floorf(fh), d0f = floorf(fd);
  const float tw = fw - w0f, th = fh - h0f, td = fd - d0f;
  const int w0 = (int)w0f, h0 = (int)h0f, d0 = (int)d0f;

  float feats[16];
#pragma unroll
  for (int i = 0; i < 16; ++i) feats[i] = 0.f;

  __syncthreads();   // MLP weights ready

#pragma unroll
  for (int dd = 0; dd < 2; ++dd)
#pragma unroll
    for (int hh = 0; hh < 2; ++hh)
#pragma unroll
      for (int ww = 0; ww < 2; ++ww) {
        const int di = d0 + dd, hi = h0 + hh, wi = w0 + ww;
        if (di < 0 || di > 15 || hi < 0 || hi > 127 || wi < 0 || wi > 127)
          continue;  // zero-padding: OOB corners contribute nothing
        const float wt = (dd ? td : 1.f - td) * (hh ? th : 1.f - th) *
                         (ww ? tw : 1.f - tw);
        if (use_nfg2) {
          const float4* p = (const float4*)(nfg2 +
              ((((size_t)bb * 16 + di) * 128 + hi) * 128 + wi) * 16);
#pragma unroll
          for (int q = 0; q < 4; ++q) {
            float4 vv = p[q];
            feats[q * 4 + 0] += wt * vv.x;
            feats[q * 4 + 1] += wt * vv.y;
            feats[q * 4 + 2] += wt * vv.z;
            feats[q * 4 + 3] += wt * vv.w;
          }
        } else {
#pragma unroll
          for (int f = 0; f < 16; ++f)
            feats[f] += wt * nfg[((((size_t)bb * 16 + f) * 16 + di) * 128 + hi)
                                  * 128 + wi];
        }
      }

  // MLP: h = relu(feats @ W2 + b2); alpha = sigmoid(h @ W3 + b3)
  float dot3 = 0.f;
#pragma unroll
  for (int j = 0; j < 32; ++j) {
    float hj = sb2[j];
#pragma unroll
    for (int k = 0; k < 16; ++k) hj = fmaf(feats[k], sW2[k * 32 + j], hj);
    hj = fmaxf(hj, 0.f);
    dot3 = fmaf(hj, sW3[j], dot3);
  }
  const float alpha = 1.0f / (1.0f + __expf(-(dot3 + sb3)));
  const float oma   = 1.0f - alpha + 1e-10f;

  // exclusive cumulative product across samples (Hillis-Steele in LDS)
  scan[t] = oma;
  __syncthreads();
  for (int off = 1; off < NSAMP; off <<= 1) {
    const float v = (t >= off) ? scan[t - off] : 1.0f;
    __syncthreads();
    scan[t] *= v;
    __syncthreads();
  }
  const float trans = (t == 0) ? 1.0f : scan[t - 1];
  const float wgt = alpha * trans;
  wout[(size_t)ray * NSAMP + t] = wgt;

  red[t] = wgt * depth;
  __syncthreads();
#pragma unroll
  for (int off = 64; off >= 1; off >>= 1) {
    if (t < off) red[t] += red[t + off];
    __syncthreads();
  }
  if (t == 0) pred[ray] = red[0];
}

// ---------------------------------------------------------------------------
extern "C" void kernel_launch(void* const* d_in, const int* in_sizes, int n_in,
                              void* d_out, int out_size, void* d_ws, size_t ws_size,
                              hipStream_t stream) {
  const float* dec  = (const float*)d_in[0];
  const float* ro   = (const float*)d_in[1];
  const float* rdir = (const float*)d_in[2];
  const float* ln_g = (const float*)d_in[3];
  const float* ln_b = (const float*)d_in[4];
  const float* W1   = (const float*)d_in[5];
  const float* b1   = (const float*)d_in[6];
  const float* W2   = (const float*)d_in[7];
  const float* b2   = (const float*)d_in[8];
  const float* W3   = (const float*)d_in[9];
  const float* b3   = (const float*)d_in[10];

  float* out     = (float*)d_out;
  float* pred    = out + PRED_OFF;
  float* weights = out + WEIGHT_OFF;
  float* nfg     = out + NFG_OFF;

  float* A_t  = (float*)d_ws;                         // 512*2048 floats (4 MB)
  float* nfg2 = A_t + (size_t)DIN * MROWS;            // 8388608 floats (32 MB)
  const size_t need = ((size_t)DIN * MROWS + (size_t)NFG_ELEMS) * sizeof(float);
  const int use_nfg2 = (ws_size >= need) ? 1 : 0;

  ln_kernel<<<MROWS / 8, 256, 0, stream>>>(dec, ln_g, ln_b, A_t);

  dim3 gg(NOUT / 64, MROWS / 32);
  gemm_nfg_kernel<<<gg, 256, 0, stream>>>(A_t, W1, b1, nfg, nfg2, use_nfg2);

  ray_kernel<<<NRAYS, NSAMP, 0, stream>>>(ro, rdir, W2, b2, W3, b3,
                                          nfg, nfg2, use_nfg2, pred, weights);
}